// ThreeStreamMaha_GCN_ModelvB_22299470200865
// MI455X (gfx1250) — compile-verified
//
#include <hip/hip_runtime.h>
#include <hip/hip_bf16.h>

// Problem shape (compile-time constants from the reference)
#define NB  256
#define NT  150
#define NJ  25
#define NC  3
#define NP  300                 // NJ*(NJ-1)/2
#define NE  (NT * NP)           // 45000 jcd values per batch
#define NG  (NE / 4)            // 11250 float4 groups (NP % 4 == 0 -> no t straddle)
#define GPT (NP / 4)            // 75 groups per frame
#define PBF (NT * NJ * NC)      // 11250 floats of p per batch
#define TPB 1024                // 32 waves (wave32)

#define EPS_F     1e-8f
#define INV_PI_F  0.31830988618379067154f
#define SPATIAL_W 0.3f
#define TEMPORAL_W 0.3f
#define ANGULAR_W 0.4f

typedef unsigned int tdm_v4u __attribute__((ext_vector_type(4)));
typedef int          tdm_v8i __attribute__((ext_vector_type(8)));
typedef int          tdm_v4i __attribute__((ext_vector_type(4)));

struct GMat { float g00, g01, g02, g11, g12, g22; };

// ||d @ L^T||^2 == d^T (L^T L) d  -> 6-term quadratic form with G = L^T L
__device__ __forceinline__ float qform(const GMat& G, float d0, float d1, float d2) {
  float q = G.g00 * d0 * d0;
  q = fmaf(G.g11 * d1, d1, q);
  q = fmaf(G.g22 * d2, d2, q);
  float c = G.g01 * d0 * d1;
  c = fmaf(G.g02 * d0, d2, c);
  c = fmaf(G.g12 * d1, d2, c);
  return fmaf(2.0f, c, q);
}

// One raw (pre-normalization) jcd value for frame t, joints (i,j), all from LDS.
__device__ __forceinline__ float jcd_pair(const float* __restrict__ row,  // p_s + t*75
                                          const GMat& G, bool t0, int i, int j) {
  const float* Pi = row + i * NC;
  const float* Pj = row + j * NC;

  const float d0 = Pi[0] - Pj[0];
  const float d1 = Pi[1] - Pj[1];
  const float d2 = Pi[2] - Pj[2];
  const float spatial = sqrtf(fmaxf(qform(G, d0, d1, d2), 0.0f));
  if (t0) return spatial;

  // velocities at absolute frame t: p[t] - p[t-1]
  const float vi0 = Pi[0] - Pi[0 - NJ * NC];
  const float vi1 = Pi[1] - Pi[1 - NJ * NC];
  const float vi2 = Pi[2] - Pi[2 - NJ * NC];
  const float vj0 = Pj[0] - Pj[0 - NJ * NC];
  const float vj1 = Pj[1] - Pj[1 - NJ * NC];
  const float vj2 = Pj[2] - Pj[2 - NJ * NC];

  const float temporal =
      sqrtf(fmaxf(qform(G, vi0 - vj0, vi1 - vj1, vi2 - vj2), 0.0f));

  const float ni  = sqrtf(fmaf(vi0, vi0, fmaf(vi1, vi1, vi2 * vi2)));
  const float nj  = sqrtf(fmaf(vj0, vj0, fmaf(vj1, vj1, vj2 * vj2)));
  const float dot = fmaf(vi0, vj0, fmaf(vi1, vj1, vi2 * vj2));
  float cs = dot / ((ni + EPS_F) * (nj + EPS_F));
  cs = fminf(1.0f, fmaxf(-1.0f, cs));
  const float angular = acosf(cs) * INV_PI_F;

  return SPATIAL_W * spatial + TEMPORAL_W * temporal + ANGULAR_W * angular;
}

// Compute one float4 group (4 consecutive pairs of one frame).
__device__ __forceinline__ float4 jcd_group(const float* __restrict__ p_s,
                                            const unsigned short* __restrict__ pair_s,
                                            const GMat& G, int g) {
  const int t  = g / GPT;                 // 75 groups per frame
  const int k0 = (g - t * GPT) * 4;
  const bool t0 = (t == 0);
  const float* row = p_s + t * (NJ * NC);

  // 4 packed (i | j<<8) pair codes in one 8-byte LDS load (k0 % 4 == 0)
  const ushort4 pk = *reinterpret_cast<const ushort4*>(pair_s + k0);

  float4 v;
  v.x = jcd_pair(row, G, t0, (int)(pk.x & 0xff), (int)(pk.x >> 8));
  v.y = jcd_pair(row, G, t0, (int)(pk.y & 0xff), (int)(pk.y >> 8));
  v.z = jcd_pair(row, G, t0, (int)(pk.z & 0xff), (int)(pk.z >> 8));
  v.w = jcd_pair(row, G, t0, (int)(pk.w & 0xff), (int)(pk.w >> 8));
  return v;
}

__global__ __launch_bounds__(TPB)
void ThreeStreamMaha_jcd_kernel(const float* __restrict__ p,
                                const float* __restrict__ L,
                                float* __restrict__ out) {
  __shared__ float p_s[PBF];                       // 45000 B: this batch's p slab
  __shared__ float redmn[32];
  __shared__ float redmx[32];
  __shared__ __align__(16) unsigned short pair_s[NP];  // packed (i | j<<8)

  const int b    = blockIdx.x;
  const int tid  = threadIdx.x;
  const int wave = tid >> 5;
  const int lane = tid & 31;

  // ---- Stage p[b] into LDS with the Tensor Data Mover (wave 0 issues once) ----
  if (wave == 0) {
#if __has_builtin(__builtin_amdgcn_tensor_load_to_lds) && __has_builtin(__builtin_amdgcn_s_wait_tensorcnt)
    const unsigned long long ga =
        (unsigned long long)(const void*)(p + (size_t)b * PBF);
    const unsigned lds_off = (unsigned)(uintptr_t)(void*)&p_s[0];

    // D# group 0: count=1 | lds_addr | global_addr[56:0] | type=2
    tdm_v4u g0;
    g0.x = 1u;                                               // count=1, user desc
    g0.y = lds_off;                                          // LDS byte address
    g0.z = (unsigned)(ga & 0xffffffffu);                     // global_addr[31:0] -> bits 95:64
    g0.w = (unsigned)((ga >> 32) & 0x01ffffffu) | (2u << 30);// addr[56:32] | type=2

    // D# group 1: 1-D tile of PBF fp32 elements, contiguous
    tdm_v8i g1;
    g1[0] = (int)(2u << 16);                // workgroup_mask=0 (not a cluster), data_size=4B
    g1[1] = (int)((unsigned)PBF << 16);     // tensor_dim0[15:0]  @bits 63:48
    g1[2] = (int)(1u << 16);                // tensor_dim0 hi=0 | tensor_dim1=1
    g1[3] = (int)((unsigned)PBF << 16);     // tile_dim0 = 11250  @bits 127:112
    g1[4] = 1;                              // tile_dim1 = 1, tile_dim2 = 0
    g1[5] = PBF;                            // tensor_dim0_stride[31:0]
    g1[6] = 0;                              // stride hi, tensor_dim1_stride lo
    g1[7] = 0;

    tdm_v4i gz4 = {0, 0, 0, 0};             // groups 2/3 unused (<=2D tensor)
    tdm_v8i gz8 = {0, 0, 0, 0, 0, 0, 0, 0}; // extra descriptor slot (clang-23 6-arg form)
    __builtin_amdgcn_tensor_load_to_lds(g0, g1, gz4, gz4, gz8, 0);
    __builtin_amdgcn_s_wait_tensorcnt(0);   // TENSORcnt==0 before the barrier
#else
    for (int i = tid; i < PBF; i += 32) p_s[i] = p[(size_t)b * PBF + i];
#endif
  }

  // ---- Pair table (i<j, triu order), packed i | j<<8 ----
  for (int k = tid; k < NP; k += TPB) {
    int i = 0, rem = k;
    while (rem >= (NJ - 1 - i)) { rem -= (NJ - 1 - i); ++i; }
    pair_s[k] = (unsigned short)(i | ((i + 1 + rem) << 8));
  }

  // ---- G = L^T L (uniform, 9 broadcast loads per thread) ----
  GMat G;
  {
    const float l00 = L[0], l01 = L[1], l02 = L[2];
    const float l10 = L[3], l11 = L[4], l12 = L[5];
    const float l20 = L[6], l21 = L[7], l22 = L[8];
    G.g00 = fmaf(l00, l00, fmaf(l10, l10, l20 * l20));
    G.g01 = fmaf(l00, l01, fmaf(l10, l11, l20 * l21));
    G.g02 = fmaf(l00, l02, fmaf(l10, l12, l20 * l22));
    G.g11 = fmaf(l01, l01, fmaf(l11, l11, l21 * l21));
    G.g12 = fmaf(l01, l02, fmaf(l11, l12, l21 * l22));
    G.g22 = fmaf(l02, l02, fmaf(l12, l12, l22 * l22));
  }

  __syncthreads();   // p_s + pair table visible to all waves

  // ---- Pass 1: per-thread min/max over 45000 jcd values (float4 groups) ----
  // Rolled on purpose: full unroll caused spills + 66KB of code. The float4
  // group already provides 4 independent sqrt/acos chains of ILP per iteration.
  float lmin =  INFINITY;
  float lmax = -INFINITY;
  #pragma unroll 1
  for (int g = tid; g < NG; g += TPB) {
    const float4 v = jcd_group(p_s, pair_s, G, g);
    lmin = fminf(lmin, fminf(fminf(v.x, v.y), fminf(v.z, v.w)));
    lmax = fmaxf(lmax, fmaxf(fmaxf(v.x, v.y), fmaxf(v.z, v.w)));
  }

  // wave32 shuffle-tree reduction, then cross-wave via LDS
  #pragma unroll
  for (int m = 16; m > 0; m >>= 1) {
    lmin = fminf(lmin, __shfl_xor(lmin, m, 32));
    lmax = fmaxf(lmax, __shfl_xor(lmax, m, 32));
  }
  if (lane == 0) { redmn[wave] = lmin; redmx[wave] = lmax; }
  __syncthreads();
  if (wave == 0) {
    float a = redmn[lane];
    float c = redmx[lane];
    #pragma unroll
    for (int m = 16; m > 0; m >>= 1) {
      a = fminf(a, __shfl_xor(a, m, 32));
      c = fmaxf(c, __shfl_xor(c, m, 32));
    }
    if (lane == 0) { redmn[0] = a; redmx[0] = c; }
  }
  __syncthreads();

  const float bmin = redmn[0];
  const float bmax = redmx[0];
  const bool  donorm = bmax > bmin;
  const float scale = donorm ? 1.0f / (bmax - bmin) : 1.0f;
  const float shift = donorm ? bmin : 0.0f;   // out = (v - shift) * scale

  // ---- Pass 2: recompute (all from LDS), normalize, b128 coalesced stores ----
  float4* __restrict__ outb = reinterpret_cast<float4*>(out + (size_t)b * NE);
  #pragma unroll 1
  for (int g = tid; g < NG; g += TPB) {
    float4 v = jcd_group(p_s, pair_s, G, g);
    v.x = (v.x - shift) * scale;
    v.y = (v.y - shift) * scale;
    v.z = (v.z - shift) * scale;
    v.w = (v.w - shift) * scale;
    outb[g] = v;
  }
}

extern "C" void kernel_launch(void* const* d_in, const int* in_sizes, int n_in,
                              void* d_out, int out_size, void* d_ws, size_t ws_size,
                              hipStream_t stream) {
  (void)in_sizes; (void)n_in; (void)d_ws; (void)ws_size; (void)out_size;
  const float* p = (const float*)d_in[0];   // (256,150,25,3) fp32
  const float* L = (const float*)d_in[1];   // (3,3) fp32
  float* out = (float*)d_out;               // (256,150,300) fp32

  ThreeStreamMaha_jcd_kernel<<<NB, TPB, 0, stream>>>(p, L, out);
}